// MotionGenerator_27943057228191
// MI455X (gfx1250) — compile-verified
//
#include <hip/hip_runtime.h>

// ---------- types ----------
typedef __attribute__((ext_vector_type(2))) float v2f;
typedef __attribute__((ext_vector_type(4))) float f32x4;
typedef __attribute__((ext_vector_type(8))) float v8f;

#define N_NODES 512
#define N_EDGES 8192
#define DIM     32
#define NOBS    64
#define ED      16384
#define OUTF    7

// workspace layout (float offsets)
#define OFF_H1    0u        // 16384
#define OFF_HF    16384u    // 16384
#define OFF_O1    32768u    // 2048
#define OFF_O     34816u    // 2048
#define OFF_Q     36864u    // 16384
#define OFF_T     53248u    // 32
#define OFF_W     53280u    // 32
#define OFF_NF    53312u    // 16384
#define OFF_RED   69696u    // 128
#define OFF_ST    69824u    // 64 (mu, rstd)
#define OFF_XG    69888u    // 16384
#define OFF_XW    86272u    // 16384
#define OFF_DINV  102656u   // 512
#define OFF_NORM  103168u   // 8192
#define OFF_DEGI  111360u   // 512 (ints)
#define OFF_QPART 111872u   // 32*16384
#define QCHUNKS   32
#define QROWS     (ED / QCHUNKS)   // 512

// ---------- small MLP layer: out[m,32] = prelu(X[m,K] @ W[K,32] + b) ----------
__global__ void embed1_kernel(const float* __restrict__ X, const float* __restrict__ W,
                              const float* __restrict__ b, const float* __restrict__ alpha,
                              float* __restrict__ H, int M, int K) {
  int idx = blockIdx.x * 256 + threadIdx.x;
  if (idx >= M * DIM) return;
  int m = idx >> 5, f = idx & 31;
  float s = b[f];
  for (int k = 0; k < K; ++k) s = fmaf(X[m * K + k], W[k * DIM + f], s);
  float a = alpha[0];
  H[idx] = s >= 0.f ? s : a * s;
}

// ---------- generic D[M,32] = A[M,32] @ B[32,32] (+bias) via V_WMMA_F32_16X16X4_F32 ----------
// one wave per 16x16 output tile; grid.x = (M/16)*2
__global__ void wmma_gemm_k32(const float* __restrict__ A, const float* __restrict__ B,
                              const float* __restrict__ bias, float* __restrict__ D) {
  int tile = blockIdx.x;
  int mt = tile >> 1;
  int nt = tile & 1;
  int lane = threadIdx.x;          // 0..31, full wave, EXEC all 1s
  int half = lane >> 4;            // 0: K pair (k0,k0+1)   1: K pair (k0+2,k0+3)
  int l16  = lane & 15;
  int n    = nt * 16 + l16;

  v8f acc = {0.f, 0.f, 0.f, 0.f, 0.f, 0.f, 0.f, 0.f};
#pragma unroll
  for (int k0 = 0; k0 < DIM; k0 += 4) {
    // A 16x4 tile (ISA 7.12.2): lanes0-15 -> K=k0,k0+1 ; lanes16-31 -> K=k0+2,k0+3
    const float* arow = A + (mt * 16 + l16) * DIM + k0 + half * 2;
    v2f a = {arow[0], arow[1]};
    // B 4x16 tile, N striped across lanes (mirror of A pattern)
    v2f bb = {B[(k0 + half * 2 + 0) * DIM + n],
              B[(k0 + half * 2 + 1) * DIM + n]};
    acc = __builtin_amdgcn_wmma_f32_16x16x4_f32(false, a, false, bb,
                                                (short)0, acc, false, false);
  }
  float badd = bias ? bias[n] : 0.f;
#pragma unroll
  for (int v = 0; v < 8; ++v) {
    int row = mt * 16 + v + half * 8;     // C/D: VGPR v -> M=v (lanes<16), M=v+8 (lanes>=16)
    D[row * DIM + n] = acc[v] + badd;
  }
}

// ---------- q = hf @ Wq : partial over row-chunks, NT streaming loads ----------
__global__ void gemv_q_partial(const float* __restrict__ hf, const float* __restrict__ Wq,
                               float* __restrict__ qpart) {
  __shared__ float sh[QROWS];
  int cb = blockIdx.x;               // 0..15 column blocks of 1024 cols
  int ch = blockIdx.y;               // 0..31 row chunks of 512 rows
  int c0 = cb * 1024 + threadIdx.x * 4;
  int r0 = ch * QROWS;
  for (int i = threadIdx.x; i < QROWS; i += 256) sh[i] = hf[r0 + i];
  __syncthreads();
  f32x4 acc = {0.f, 0.f, 0.f, 0.f};
#pragma unroll 4
  for (int r = 0; r < QROWS; ++r) {
    float hv = sh[r];
    const f32x4* p = (const f32x4*)(Wq + (size_t)(r0 + r) * ED + c0);
    f32x4 wv = __builtin_nontemporal_load(p);
    acc += wv * hv;
  }
  *(f32x4*)(qpart + (size_t)ch * ED + c0) = acc;
}

__global__ void gemv_q_reduce(const float* __restrict__ qpart, float* __restrict__ q) {
  int c = blockIdx.x * 256 + threadIdx.x;
  float s = 0.f;
#pragma unroll
  for (int ch = 0; ch < QCHUNKS; ++ch) s += qpart[(size_t)ch * ED + c];
  q[c] = s;
}

// ---------- t[d] = sum_c Wk[d,c] * q[c] ----------
__global__ void gemv_t_kernel(const float* __restrict__ Wk, const float* __restrict__ q,
                              float* __restrict__ t) {
  __shared__ float red[256];
  int d = blockIdx.x;
  float s = 0.f;
  for (int c = threadIdx.x; c < ED; c += 256) s = fmaf(Wk[(size_t)d * ED + c], q[c], s);
  red[threadIdx.x] = s; __syncthreads();
  for (int w = 128; w > 0; w >>= 1) {
    if (threadIdx.x < w) red[threadIdx.x] += red[threadIdx.x + w];
    __syncthreads();
  }
  if (threadIdx.x == 0) t[d] = red[0];
}

// ---------- att[j] = o[j]·t ; w[d] = sum_j att[j]*o[j,d] ----------
__global__ void attw_kernel(const float* __restrict__ o, const float* __restrict__ t,
                            float* __restrict__ w) {
  __shared__ float att[NOBS];
  int tid = threadIdx.x;
  if (tid < NOBS) {
    float s = 0.f;
    for (int d = 0; d < DIM; ++d) s = fmaf(o[tid * DIM + d], t[d], s);
    att[tid] = s;
  }
  __syncthreads();
  if (tid < DIM) {
    float s = 0.f;
    for (int j = 0; j < NOBS; ++j) s = fmaf(att[j], o[j * DIM + tid], s);
    w[tid] = s;
  }
}

// ---------- nf[c] = w·Wv[:,c] + hf[c] ; per-block (sum, sumsq) partials ----------
__global__ void nf_kernel(const float* __restrict__ w, const float* __restrict__ Wv,
                          const float* __restrict__ hf, float* __restrict__ nf,
                          float* __restrict__ red) {
  __shared__ float s1[256], s2[256];
  int c = blockIdx.x * 256 + threadIdx.x;
  float s = hf[c];
#pragma unroll
  for (int d = 0; d < DIM; ++d) s = fmaf(w[d], Wv[(size_t)d * ED + c], s);
  nf[c] = s;
  s1[threadIdx.x] = s; s2[threadIdx.x] = s * s; __syncthreads();
  for (int k = 128; k > 0; k >>= 1) {
    if (threadIdx.x < k) { s1[threadIdx.x] += s1[threadIdx.x + k]; s2[threadIdx.x] += s2[threadIdx.x + k]; }
    __syncthreads();
  }
  if (threadIdx.x == 0) { red[blockIdx.x] = s1[0]; red[64 + blockIdx.x] = s2[0]; }
}

__global__ void stats_kernel(const float* __restrict__ red, float* __restrict__ st) {
  __shared__ float a[64], b[64];
  int tid = threadIdx.x;
  a[tid] = red[tid]; b[tid] = red[64 + tid]; __syncthreads();
  for (int k = 32; k > 0; k >>= 1) {
    if (tid < k) { a[tid] += a[tid + k]; b[tid] += b[tid + k]; }
    __syncthreads();
  }
  if (tid == 0) {
    float mu  = a[0] * (1.f / ED);
    float var = b[0] * (1.f / ED) - mu * mu;
    st[0] = mu;
    st[1] = rsqrtf(var + 1e-6f);
  }
}

__global__ void apply_ln_kernel(const float* __restrict__ nf, const float* __restrict__ st,
                                const float* __restrict__ g, const float* __restrict__ b,
                                float* __restrict__ xg) {
  int c = blockIdx.x * 256 + threadIdx.x;
  xg[c] = fmaf((nf[c] - st[0]) * st[1], g[c], b[c]);
}

// ---------- GCN degree / norm ----------
__global__ void deg_init_kernel(int* __restrict__ deg) {
  int i = blockIdx.x * 256 + threadIdx.x;
  if (i < N_NODES) deg[i] = 1;   // self loop
}
__global__ void deg_count_kernel(const int* __restrict__ dst, int* __restrict__ deg) {
  int e = blockIdx.x * 256 + threadIdx.x;
  if (e < N_EDGES) atomicAdd(&deg[dst[e]], 1);
}
__global__ void dinv_kernel(const int* __restrict__ deg, float* __restrict__ dinv) {
  int i = blockIdx.x * 256 + threadIdx.x;
  if (i < N_NODES) dinv[i] = rsqrtf((float)deg[i]);
}
__global__ void norm_kernel(const int* __restrict__ src, const int* __restrict__ dst,
                            const float* __restrict__ dinv, float* __restrict__ norm) {
  int e = blockIdx.x * 256 + threadIdx.x;
  if (e < N_EDGES) norm[e] = dinv[src[e]] * dinv[dst[e]];
}

// ---------- GCN aggregate (deterministic scan) + PReLU + final linear ----------
__global__ void gcn_final_kernel(const int* __restrict__ src, const int* __restrict__ dst,
                                 const float* __restrict__ norm, const float* __restrict__ xw,
                                 const float* __restrict__ dinv, const float* __restrict__ b_gcn,
                                 const float* __restrict__ a_gcn, const float* __restrict__ W_fin,
                                 const float* __restrict__ b_fin, float* __restrict__ out) {
  __shared__ float red[8][DIM];
  int n    = blockIdx.x;
  int lane = threadIdx.x & 31;
  int wv   = threadIdx.x >> 5;               // 8 waves, disjoint edge ranges
  float acc = 0.f;
  int e0 = wv * (N_EDGES / 8), e1 = e0 + (N_EDGES / 8);
  for (int e = e0; e < e1; ++e) {
    if (dst[e] == n) {
      int s = src[e];
      acc = fmaf(xw[s * DIM + lane], norm[e], acc);
    }
  }
  red[wv][lane] = acc; __syncthreads();
  if (wv == 0) {
    float s = red[0][lane];
#pragma unroll
    for (int i = 1; i < 8; ++i) s += red[i][lane];   // fixed order: deterministic
    float di = dinv[n];
    s = fmaf(xw[n * DIM + lane], di * di, s);        // self loop
    s += b_gcn[lane];
    float a = a_gcn[0];
    red[0][lane] = s >= 0.f ? s : a * s;             // PReLU
  }
  __syncthreads();
  if (threadIdx.x < OUTF) {
    float s = b_fin[threadIdx.x];
    for (int f = 0; f < DIM; ++f) s = fmaf(red[0][f], W_fin[f * OUTF + threadIdx.x], s);
    out[n * OUTF + threadIdx.x] = s;
  }
}

// ---------- launch ----------
extern "C" void kernel_launch(void* const* d_in, const int* in_sizes, int n_in,
                              void* d_out, int out_size, void* d_ws, size_t ws_size,
                              hipStream_t stream) {
  const float* x     = (const float*)d_in[0];
  const float* obs   = (const float*)d_in[1];
  const int*   eidx  = (const int*)  d_in[2];
  const float* W_ne1 = (const float*)d_in[3];
  const float* b_ne1 = (const float*)d_in[4];
  const float* a_ne  = (const float*)d_in[5];
  const float* W_ne2 = (const float*)d_in[6];
  const float* b_ne2 = (const float*)d_in[7];
  const float* W_oe1 = (const float*)d_in[8];
  const float* b_oe1 = (const float*)d_in[9];
  const float* a_oe  = (const float*)d_in[10];
  const float* W_oe2 = (const float*)d_in[11];
  const float* b_oe2 = (const float*)d_in[12];
  const float* Wk    = (const float*)d_in[13];
  const float* Wq    = (const float*)d_in[14];
  const float* Wv    = (const float*)d_in[15];
  const float* ln_g  = (const float*)d_in[16];
  const float* ln_b  = (const float*)d_in[17];
  const float* W_gcn = (const float*)d_in[18];
  const float* b_gcn = (const float*)d_in[19];
  const float* a_gcn = (const float*)d_in[20];
  const float* W_fin = (const float*)d_in[21];
  const float* b_fin = (const float*)d_in[22];
  float* out = (float*)d_out;

  float* ws = (float*)d_ws;
  float* h1    = ws + OFF_H1;
  float* hf    = ws + OFF_HF;
  float* o1    = ws + OFF_O1;
  float* o     = ws + OFF_O;
  float* q     = ws + OFF_Q;
  float* t     = ws + OFF_T;
  float* w     = ws + OFF_W;
  float* nf    = ws + OFF_NF;
  float* red   = ws + OFF_RED;
  float* st    = ws + OFF_ST;
  float* xg    = ws + OFF_XG;
  float* xw    = ws + OFF_XW;
  float* dinv  = ws + OFF_DINV;
  float* norm  = ws + OFF_NORM;
  int*   degi  = (int*)(ws + OFF_DEGI);
  float* qpart = ws + OFF_QPART;

  const int* esrc = eidx;            // edge_index[0]
  const int* edst = eidx + N_EDGES;  // edge_index[1]

  // 1. node / obstacle embeddings (layer 1 + PReLU)
  embed1_kernel<<<(N_NODES * DIM + 255) / 256, 256, 0, stream>>>(x, W_ne1, b_ne1, a_ne, h1, N_NODES, 7);
  embed1_kernel<<<(NOBS * DIM + 255) / 256, 256, 0, stream>>>(obs, W_oe1, b_oe1, a_oe, o1, NOBS, 6);
  // 2. layer-2 linears via fp32 WMMA
  wmma_gemm_k32<<<(N_NODES / 16) * 2, 32, 0, stream>>>(h1, W_ne2, b_ne2, hf);
  wmma_gemm_k32<<<(NOBS / 16) * 2, 32, 0, stream>>>(o1, W_oe2, b_oe2, o);
  // 3. q = hf @ Wq  (the 1 GiB stream — roofline-dominant)
  gemv_q_partial<<<dim3(ED / 1024, QCHUNKS), 256, 0, stream>>>(hf, Wq, qpart);
  gemv_q_reduce<<<ED / 256, 256, 0, stream>>>(qpart, q);
  // 4. collapsed attention: t = Wk@q ; att = o·t ; w = att@o
  gemv_t_kernel<<<DIM, 256, 0, stream>>>(Wk, q, t);
  attw_kernel<<<1, 64, 0, stream>>>(o, t, w);
  // 5. nf = w@Wv + hf ; LayerNorm
  nf_kernel<<<ED / 256, 256, 0, stream>>>(w, Wv, hf, nf, red);
  stats_kernel<<<1, 64, 0, stream>>>(red, st);
  apply_ln_kernel<<<ED / 256, 256, 0, stream>>>(nf, st, ln_g, ln_b, xg);
  // 6. GCN normalization terms
  deg_init_kernel<<<2, 256, 0, stream>>>(degi);
  deg_count_kernel<<<N_EDGES / 256, 256, 0, stream>>>(edst, degi);
  dinv_kernel<<<2, 256, 0, stream>>>(degi, dinv);
  norm_kernel<<<N_EDGES / 256, 256, 0, stream>>>(esrc, edst, dinv, norm);
  // 7. xw = xg @ W_gcn via WMMA (bias applied after aggregation)
  wmma_gemm_k32<<<(N_NODES / 16) * 2, 32, 0, stream>>>(xg, W_gcn, nullptr, xw);
  // 8. aggregate + bias + PReLU + final linear
  gcn_final_kernel<<<N_NODES, 256, 0, stream>>>(esrc, edst, norm, xw, dinv, b_gcn, a_gcn,
                                                W_fin, b_fin, out);
}